// HetGCN_11501922419052
// MI455X (gfx1250) — compile-verified
//
#include <hip/hip_runtime.h>

#define T_ 7
#define D_ 7
#define K_ 10
#define O_ 32
#define HET 56           // (T_+1)*D_
#define HETP 58          // padded LDS row stride (even -> 8B-aligned float2, conflict-free)
#define BLK 192          // 6 waves of 32; keeps static LDS < 64KB
#define WAVES (BLK / 32)

typedef __attribute__((ext_vector_type(2))) float v2f;
typedef __attribute__((ext_vector_type(8))) float v8f;

__device__ __forceinline__ float leaky_(float x) { return x >= 0.0f ? x : 0.01f * x; }
__device__ __forceinline__ float sigmoid_(float x) { return 1.0f / (1.0f + expf(-x)); }

__global__ __launch_bounds__(BLK) void hetgcn_main(
    const float* __restrict__ x_node,   // [N, 7]
    const float* __restrict__ x_het,    // [7, N, 10, 7]
    const int* __restrict__ ntype,      // [N]
    const float* __restrict__ Wc,       // [7, 7, 7]
    const float* __restrict__ bc,       // [7, 7]
    const float* __restrict__ Wa,       // [32, 56]
    const float* __restrict__ ba,       // [32]
    float* __restrict__ blockPart,      // [gridDim.x, 32]
    int N) {
  __shared__ float sWc[T_ * D_ * D_];
  __shared__ float sbc[T_ * D_];
  __shared__ float sWa[O_ * HET];
  __shared__ float sba[O_];
  __shared__ float sHet[BLK * HETP];
  __shared__ float sPart[WAVES * O_];

  const int tid = threadIdx.x;
  for (int i = tid; i < T_ * D_ * D_; i += BLK) sWc[i] = Wc[i];
  for (int i = tid; i < T_ * D_; i += BLK) sbc[i] = bc[i];
  for (int i = tid; i < O_ * HET; i += BLK) sWa[i] = Wa[i];
  if (tid < O_) sba[tid] = ba[tid];
  __syncthreads();

  const int node = blockIdx.x * BLK + tid;
  const bool valid = node < N;

  float het[HET];
#pragma unroll
  for (int i = 0; i < HET; ++i) het[i] = 0.0f;

  if (valid) {
    float buf[K_ * D_];
    for (int t = 0; t < T_; ++t) {
      const float* base = x_het + ((size_t)t * N + node) * (K_ * D_);
      if (t + 1 < T_)
        __builtin_prefetch(x_het + ((size_t)(t + 1) * N + node) * (K_ * D_), 0, 1);
      // 70 contiguous floats, base always 8B-aligned ((t*N+n)*280 bytes)
#pragma unroll
      for (int i = 0; i < (K_ * D_) / 2; ++i) {
        v2f v = ((const v2f*)base)[i];
        buf[2 * i] = v.x;
        buf[2 * i + 1] = v.y;
      }
      float acc[D_];
#pragma unroll
      for (int o = 0; o < D_; ++o) acc[o] = 0.0f;
      int cnt = 0;
#pragma unroll
      for (int k = 0; k < K_; ++k) {
        bool nz = false;
        float e[D_];
#pragma unroll
        for (int o = 0; o < D_; ++o) {
          float s = sbc[t * D_ + o];
#pragma unroll
          for (int d = 0; d < D_; ++d)
            s = fmaf(buf[k * D_ + d], sWc[(t * D_ + o) * D_ + d], s);
          s = leaky_(s);
          e[o] = s;
          nz = nz || (s != 0.0f);
        }
        if (nz) cnt++;
        // masked rows are all-zero after leaky_relu, so plain sum == masked sum
#pragma unroll
        for (int o = 0; o < D_; ++o) acc[o] += e[o];
      }
      const float cf = (float)(cnt > 0 ? cnt : 1);
#pragma unroll
      for (int o = 0; o < D_; ++o) het[t * D_ + o] = acc[o] / cf;
    }
    // self embedding with node's own type
    const int ty = ntype[node];
#pragma unroll
    for (int o = 0; o < D_; ++o) {
      float s = sbc[ty * D_ + o];
#pragma unroll
      for (int d = 0; d < D_; ++d)
        s = fmaf(x_node[(size_t)node * D_ + d], sWc[(ty * D_ + o) * D_ + d], s);
      het[T_ * D_ + o] = leaky_(s);
    }
  }

  // stage het rows for WMMA (zeros for ragged-tail rows)
#pragma unroll
  for (int i = 0; i < HET; ++i) sHet[tid * HETP + i] = het[i];
  __syncthreads();

  // --- het[BLK,56] x W_agg^T[56,32] via V_WMMA_F32_16X16X4_F32, EXEC all-ones here ---
  const int lane = tid & 31;
  const int wave = tid >> 5;
  const int laneLo = lane & 15;
  const bool hi = lane >= 16;
  const int blockBase = blockIdx.x * BLK;

#pragma unroll
  for (int ct = 0; ct < 2; ++ct) {
    const int col = ct * 16 + laneLo;   // output column 0..31
    const float bias = sba[col];
    float colsum = 0.0f;
#pragma unroll
    for (int rt = 0; rt < 2; ++rt) {
      const int rowBase = wave * 32 + rt * 16;
      const float* arow = &sHet[(size_t)(rowBase + laneLo) * HETP];
      v8f c = {0.0f, 0.0f, 0.0f, 0.0f, 0.0f, 0.0f, 0.0f, 0.0f};
#pragma unroll
      for (int kk = 0; kk < 14; ++kk) {  // K = 56 = 14 * 4, exact
        const int k0 = 4 * kk + (hi ? 2 : 0);
        // A 16x4 f32: lanes0-15 V0/V1 = K0/K1; lanes16-31 V0/V1 = K2/K3 (ISA 7.12.2)
        v2f a = *(const v2f*)&arow[k0];
        // B 4x16 f32: B[k][col] = W_agg[col][k]; K rows striped per half-wave
        v2f b;
        b.x = sWa[col * HET + k0];
        b.y = sWa[col * HET + k0 + 1];
        c = __builtin_amdgcn_wmma_f32_16x16x4_f32(false, a, false, b, (short)0, c,
                                                  false, false);
      }
      const int rOff = hi ? 8 : 0;  // C/D: lanes16-31 hold rows M=8..15
#pragma unroll
      for (int r = 0; r < 8; ++r) {
        if (blockBase + rowBase + rOff + r < N)
          colsum += sigmoid_(c[r] + bias);
      }
    }
    colsum += __shfl_xor(colsum, 16, 32);  // combine row halves (same column)
    if (!hi) sPart[wave * O_ + col] = colsum;
  }
  __syncthreads();

  if (tid < O_) {
    float s = 0.0f;
#pragma unroll
    for (int w = 0; w < WAVES; ++w) s += sPart[w * O_ + tid];
    blockPart[(size_t)blockIdx.x * O_ + tid] = s;  // deterministic per-block partial
  }
}

__global__ void hetgcn_finalize(const float* __restrict__ blockPart, int nBlocks,
                                float* __restrict__ out, int N) {
  const int o = threadIdx.x;
  if (o < O_) {
    float s = 0.0f;
    for (int b = 0; b < nBlocks; ++b) s += blockPart[(size_t)b * O_ + o];
    out[o] = s / (float)N;
  }
}

extern "C" void kernel_launch(void* const* d_in, const int* in_sizes, int n_in,
                              void* d_out, int out_size, void* d_ws, size_t ws_size,
                              hipStream_t stream) {
  const float* x_node = (const float*)d_in[0];
  const float* x_het = (const float*)d_in[1];
  const int* ntype = (const int*)d_in[2];
  const float* Wc = (const float*)d_in[3];
  const float* bc = (const float*)d_in[4];
  const float* Wa = (const float*)d_in[5];
  const float* ba = (const float*)d_in[6];
  float* out = (float*)d_out;

  const int N = in_sizes[2];  // graph_node_types has N elements
  const int nBlocks = (N + BLK - 1) / BLK;
  float* blockPart = (float*)d_ws;  // nBlocks * 32 floats

  hipLaunchKernelGGL(hetgcn_main, dim3(nBlocks), dim3(BLK), 0, stream,
                     x_node, x_het, ntype, Wc, bc, Wa, ba, blockPart, N);
  hipLaunchKernelGGL(hetgcn_finalize, dim3(1), dim3(64), 0, stream,
                     blockPart, nBlocks, out, N);
}